// GCN_attack_70411693850860
// MI455X (gfx1250) — compile-verified
//
#include <hip/hip_runtime.h>
#include <cstdint>

typedef float v2f __attribute__((ext_vector_type(2)));
typedef float v8f __attribute__((ext_vector_type(8)));

#define NFEAT 256
#define NHID  128
#define NCLASS 41
#define NCPAD  48

// ---------------------------------------------------------------- degree ----
__global__ __launch_bounds__(256) void k_deg_init(float* deg, int N) {
    int i = blockIdx.x * 256 + threadIdx.x;
    if (i < N) deg[i] = 1.0f;                       // self-loop weight
}

__global__ __launch_bounds__(256) void k_deg_edges(const int* __restrict__ dst,
                                                   const float* __restrict__ w,
                                                   float* deg, int E) {
    int e = blockIdx.x * 256 + threadIdx.x;
    if (e < E) unsafeAtomicAdd(&deg[dst[e]], w[e]);
}

__global__ __launch_bounds__(256) void k_deg_finish(float* dinv, int N) {
    int i = blockIdx.x * 256 + threadIdx.x;
    if (i < N) {
        float d = dinv[i];
        dinv[i] = d > 0.0f ? rsqrtf(d) : 0.0f;
    }
}

// --------------------------------------------------------- weight repack ----
// Wp layout: [kpair][n] of float2 {W[k][n], W[k+1][n]}  (B-fragment friendly)
__global__ __launch_bounds__(256) void k_pack_w1(const float* __restrict__ W1, float* __restrict__ Wp) {
    int idx = blockIdx.x * 256 + threadIdx.x;       // 256*128 elements
    if (idx >= NFEAT * NHID) return;
    int k = idx >> 7, n = idx & 127;
    Wp[(size_t)(k >> 1) * (NHID * 2) + n * 2 + (k & 1)] = W1[idx];
}

__global__ __launch_bounds__(256) void k_pack_w2(const float* __restrict__ W2, float* __restrict__ Wp) {
    int idx = blockIdx.x * 256 + threadIdx.x;       // 128*48 elements
    if (idx >= NHID * NCPAD) return;
    int k = idx / NCPAD, n = idx % NCPAD;
    float v = (n < NCLASS) ? W2[k * NCLASS + n] : 0.0f;
    Wp[(size_t)(k >> 1) * (NCPAD * 2) + n * 2 + (k & 1)] = v;
}

// ------------------------------------------------------- GEMM1: X @ W1 ------
// Block: 256 thr = 8 waves. M-tile 32, N = 128 (wave w owns n0 = 16*w, 2 m-subtiles).
__global__ __launch_bounds__(256) void k_gemm1(const float* __restrict__ X,
                                               const float* __restrict__ Wp,
                                               float* __restrict__ H, int M) {
    __shared__ float2 Xs[128 * 33];                 // [kpair][m(32)+pad]
    const int tid = threadIdx.x;
    const int m0 = blockIdx.x * 32;

    // stage 32 rows x 256 cols, K-pair-major (coalesced float4 global reads)
#pragma unroll
    for (int it = 0; it < 8; ++it) {
        int q = it * 256 + tid;                     // 2048 float4
        int m = q >> 6, k4 = q & 63;
        int mr = m0 + m; if (mr >= M) mr = M - 1;
        float4 v = ((const float4*)(X + (size_t)mr * NFEAT))[k4];
        Xs[(2 * k4)     * 33 + m] = make_float2(v.x, v.y);
        Xs[(2 * k4 + 1) * 33 + m] = make_float2(v.z, v.w);
    }
    __syncthreads();

    const int wave = tid >> 5, lane = tid & 31, l = lane & 15, half = lane >> 4;
    const int n0 = wave * 16;
    const float2* Bp = (const float2*)Wp;           // [128 kp][128 n]

    v8f acc0 = {}, acc1 = {};
#pragma unroll 4
    for (int kp = 0; kp < 128; kp += 2) {           // K step 4
        int row = kp + half;
        float2 a0 = Xs[row * 33 + l];
        float2 a1 = Xs[row * 33 + 16 + l];
        float2 b  = Bp[(size_t)row * NHID + n0 + l];
        v2f av0 = {a0.x, a0.y}, av1 = {a1.x, a1.y}, bv = {b.x, b.y};
        acc0 = __builtin_amdgcn_wmma_f32_16x16x4_f32(false, av0, false, bv, (short)0, acc0, false, false);
        acc1 = __builtin_amdgcn_wmma_f32_16x16x4_f32(false, av1, false, bv, (short)0, acc1, false, false);
    }

    const int mb = 8 * half;
    float* Ho = H + (size_t)(m0 + mb) * NHID + n0 + l;
    if (m0 + 32 <= M) {                             // full tile: straight-line stores
#pragma unroll
        for (int r = 0; r < 8; ++r) {
            Ho[(size_t)r * NHID]        = acc0[r];
            Ho[(size_t)(16 + r) * NHID] = acc1[r];
        }
    } else {                                        // ragged tail fallback
#pragma unroll
        for (int r = 0; r < 8; ++r) {
            if (m0 + mb + r < M)      Ho[(size_t)r * NHID]        = acc0[r];
            if (m0 + 16 + mb + r < M) Ho[(size_t)(16 + r) * NHID] = acc1[r];
        }
    }
}

// ------------------------------------------- layer-1 aggregation ------------
__global__ __launch_bounds__(256) void k_agg1_init(const float* __restrict__ H,
                                                   const float* __restrict__ dinv,
                                                   float* __restrict__ Hacc, int N) {
    int idx = blockIdx.x * 256 + threadIdx.x;       // N*32 float4
    if (idx >= N * 32) return;
    int i = idx >> 5, j = idx & 31;
    float di = dinv[i], c = di * di;                // self-loop norm
    float4 v = ((const float4*)(H + (size_t)i * NHID))[j];
    float4 o = make_float4(c * v.x, c * v.y, c * v.z, c * v.w);
    ((float4*)(Hacc + (size_t)i * NHID))[j] = o;
}

__global__ __launch_bounds__(256) void k_agg1_edges(const int* __restrict__ src,
                                                    const int* __restrict__ dst,
                                                    const float* __restrict__ w,
                                                    const float* __restrict__ dinv,
                                                    const float* __restrict__ H,
                                                    float* __restrict__ Hacc, int E) {
    int e = blockIdx.x * 8 + (threadIdx.x >> 5);    // one wave per edge
    if (e >= E) return;
    int lane = threadIdx.x & 31;
    int s = src[e], d = dst[e];
    float c = dinv[s] * w[e] * dinv[d];
    float4 v = ((const float4*)(H + (size_t)s * NHID))[lane];
    float* o = Hacc + (size_t)d * NHID + lane * 4;
    unsafeAtomicAdd(o + 0, c * v.x);
    unsafeAtomicAdd(o + 1, c * v.y);
    unsafeAtomicAdd(o + 2, c * v.z);
    unsafeAtomicAdd(o + 3, c * v.w);
}

// --------------------------------------- GEMM2: relu(Hacc+b1) @ W2 ----------
// Block: 64 thr = 2 waves. M-tile 32 (wave w -> m-subtile w), N = 48 (3 n-tiles).
__global__ __launch_bounds__(64) void k_gemm2(const float* __restrict__ Hacc,
                                              const float* __restrict__ b1,
                                              const float* __restrict__ Wp,
                                              float* __restrict__ G, int M) {
    __shared__ float2 Xs[64 * 33];                  // [kpair(64)][m(32)+pad]
    const int tid = threadIdx.x;
    const int m0 = blockIdx.x * 32;

#pragma unroll
    for (int it = 0; it < 16; ++it) {
        int q = it * 64 + tid;                      // 1024 float4
        int m = q >> 5, k4 = q & 31;
        int mr = m0 + m; if (mr >= M) mr = M - 1;
        float4 v = ((const float4*)(Hacc + (size_t)mr * NHID))[k4];
        float4 bb = ((const float4*)b1)[k4];
        v.x = fmaxf(v.x + bb.x, 0.0f);              // fused bias + ReLU
        v.y = fmaxf(v.y + bb.y, 0.0f);
        v.z = fmaxf(v.z + bb.z, 0.0f);
        v.w = fmaxf(v.w + bb.w, 0.0f);
        Xs[(2 * k4)     * 33 + m] = make_float2(v.x, v.y);
        Xs[(2 * k4 + 1) * 33 + m] = make_float2(v.z, v.w);
    }
    __syncthreads();

    const int wave = tid >> 5, lane = tid & 31, l = lane & 15, half = lane >> 4;
    const float2* Bp = (const float2*)Wp;           // [64 kp][48 n]

    v8f acc0 = {}, acc1 = {}, acc2 = {};
#pragma unroll 4
    for (int kp = 0; kp < 64; kp += 2) {
        int row = kp + half;
        float2 a  = Xs[row * 33 + wave * 16 + l];
        float2 b0 = Bp[(size_t)row * NCPAD + 0  + l];
        float2 b1f= Bp[(size_t)row * NCPAD + 16 + l];
        float2 b2f= Bp[(size_t)row * NCPAD + 32 + l];
        v2f av = {a.x, a.y};
        v2f bv0 = {b0.x, b0.y}, bv1 = {b1f.x, b1f.y}, bv2 = {b2f.x, b2f.y};
        acc0 = __builtin_amdgcn_wmma_f32_16x16x4_f32(false, av, false, bv0, (short)0, acc0, false, false);
        acc1 = __builtin_amdgcn_wmma_f32_16x16x4_f32(false, av, false, bv1, (short)0, acc1, false, false);
        acc2 = __builtin_amdgcn_wmma_f32_16x16x4_f32(false, av, false, bv2, (short)0, acc2, false, false);
    }

    const int mb = m0 + wave * 16 + 8 * half;
    float* Go = G + (size_t)mb * NCPAD + l;
    if (m0 + 32 <= M) {                             // full tile: straight-line stores
#pragma unroll
        for (int r = 0; r < 8; ++r) {
            float* o = Go + (size_t)r * NCPAD;
            o[0]  = acc0[r];
            o[16] = acc1[r];
            o[32] = acc2[r];
        }
    } else {
#pragma unroll
        for (int r = 0; r < 8; ++r) {
            if (mb + r < M) {
                float* o = Go + (size_t)r * NCPAD;
                o[0]  = acc0[r];
                o[16] = acc1[r];
                o[32] = acc2[r];
            }
        }
    }
}

// ------------------------------------------- layer-2 aggregation ------------
__global__ __launch_bounds__(256) void k_agg2_init(const float* __restrict__ G,
                                                   const float* __restrict__ dinv,
                                                   float* __restrict__ Oacc, int N) {
    int idx = blockIdx.x * 256 + threadIdx.x;       // N*12 float4
    if (idx >= N * 12) return;
    int i = idx / 12, j = idx % 12;
    float di = dinv[i], c = di * di;
    float4 v = ((const float4*)(G + (size_t)i * NCPAD))[j];
    float4 o = make_float4(c * v.x, c * v.y, c * v.z, c * v.w);
    ((float4*)(Oacc + (size_t)i * NCPAD))[j] = o;
}

__global__ __launch_bounds__(256) void k_agg2_edges(const int* __restrict__ src,
                                                    const int* __restrict__ dst,
                                                    const float* __restrict__ w,
                                                    const float* __restrict__ dinv,
                                                    const float* __restrict__ G,
                                                    float* __restrict__ Oacc, int E) {
    int e = blockIdx.x * 8 + (threadIdx.x >> 5);    // one wave per edge
    if (e >= E) return;
    int lane = threadIdx.x & 31;
    int s = src[e], d = dst[e];
    float c = dinv[s] * w[e] * dinv[d];
    const float* g = G + (size_t)s * NCPAD;
    float* o = Oacc + (size_t)d * NCPAD;
    unsafeAtomicAdd(o + lane, c * g[lane]);
    if (lane < 16) unsafeAtomicAdd(o + 32 + lane, c * g[32 + lane]);
}

// --------------------------------------------- bias + log_softmax -----------
__global__ __launch_bounds__(256) void k_logsoftmax(const float* __restrict__ Oacc,
                                                    const float* __restrict__ b2,
                                                    float* __restrict__ out, int N) {
    int i = blockIdx.x * 8 + (threadIdx.x >> 5);    // one wave per row
    if (i >= N) return;
    int lane = threadIdx.x & 31;
    const float* row = Oacc + (size_t)i * NCPAD;

    float v1 = row[lane] + b2[lane];                // lane < 32 <= 41: valid
    int c2 = 32 + lane;
    bool has2 = c2 < NCLASS;
    float v2 = has2 ? (row[c2] + b2[c2]) : -INFINITY;

    float mx = fmaxf(v1, v2);
#pragma unroll
    for (int off = 16; off >= 1; off >>= 1) mx = fmaxf(mx, __shfl_xor(mx, off));

    float s = expf(v1 - mx) + (has2 ? expf(v2 - mx) : 0.0f);
#pragma unroll
    for (int off = 16; off >= 1; off >>= 1) s += __shfl_xor(s, off);

    float ls = logf(s);
    out[(size_t)i * NCLASS + lane] = v1 - mx - ls;
    if (has2) out[(size_t)i * NCLASS + c2] = v2 - mx - ls;
}

// ---------------------------------------------------------------- launch ----
extern "C" void kernel_launch(void* const* d_in, const int* in_sizes, int n_in,
                              void* d_out, int out_size, void* d_ws, size_t ws_size,
                              hipStream_t stream) {
    const float* x  = (const float*)d_in[0];
    const int*   ei = (const int*)  d_in[1];
    const float* ew = (const float*)d_in[2];
    const float* W1 = (const float*)d_in[3];
    const float* b1 = (const float*)d_in[4];
    const float* W2 = (const float*)d_in[5];
    const float* b2 = (const float*)d_in[6];
    float* out = (float*)d_out;

    const int N = in_sizes[0] / NFEAT;
    const int E = in_sizes[2];
    const int* src = ei;
    const int* dst = ei + E;

    // workspace carve-up (floats, 1KB-aligned chunks)
    float* ws = (float*)d_ws;
    size_t o = 0;
    auto take = [&](size_t n) { float* p = ws + o; o += (n + 255) & ~(size_t)255; return p; };
    float* dinv = take((size_t)N);
    float* Wp1  = take((size_t)NFEAT * NHID);
    float* Wp2  = take((size_t)NHID * NCPAD);
    float* Hxw  = take((size_t)N * NHID);
    float* Hacc = take((size_t)N * NHID);
    float* G    = Hxw;                              // reuse: Hxw dead after agg1
    float* Oacc = Hxw + (size_t)N * NCPAD;          // fits inside Hxw region

    k_pack_w1<<<(NFEAT * NHID + 255) / 256, 256, 0, stream>>>(W1, Wp1);
    k_pack_w2<<<(NHID * NCPAD + 255) / 256, 256, 0, stream>>>(W2, Wp2);

    k_deg_init  <<<(N + 255) / 256, 256, 0, stream>>>(dinv, N);
    k_deg_edges <<<(E + 255) / 256, 256, 0, stream>>>(dst, ew, dinv, E);
    k_deg_finish<<<(N + 255) / 256, 256, 0, stream>>>(dinv, N);

    k_gemm1<<<(N + 31) / 32, 256, 0, stream>>>(x, Wp1, Hxw, N);

    k_agg1_init <<<(N * 32 + 255) / 256, 256, 0, stream>>>(Hxw, dinv, Hacc, N);
    k_agg1_edges<<<(E + 7) / 8, 256, 0, stream>>>(src, dst, ew, dinv, Hxw, Hacc, E);

    k_gemm2<<<(N + 31) / 32, 64, 0, stream>>>(Hacc, b1, Wp2, G, N);

    k_agg2_init <<<(N * 12 + 255) / 256, 256, 0, stream>>>(G, dinv, Oacc, N);
    k_agg2_edges<<<(E + 7) / 8, 256, 0, stream>>>(src, dst, ew, dinv, G, Oacc, E);

    k_logsoftmax<<<(N + 7) / 8, 256, 0, stream>>>(Oacc, b2, out, N);
}